// DeepHOTCO_v4_22643067585221
// MI455X (gfx1250) — compile-verified
//
#include <hip/hip_runtime.h>
#include <cstdint>
#include <cstddef>

#define NDIM 17
#define NN   (NDIM * NDIM)   // 289 dwords per matrix
#define WPAD 292             // padded to multiple of 4 dwords -> 16B-aligned rows in LDS
#define WG   64

__device__ __forceinline__ float fast_rcp(float x) {
  return __builtin_amdgcn_rcpf(x);     // v_rcp_f32, ~1 ulp
}

__device__ __forceinline__ float sigmoid_f(float x) {
  return fast_rcp(1.0f + __expf(-x));
}

__global__ void __launch_bounds__(WG)
grossberg_scan_kernel(const float* __restrict__ state0,
                      const float* __restrict__ W_pos,
                      const float* __restrict__ W_neg,
                      const float* __restrict__ feas,
                      const float* __restrict__ pert,
                      const float* __restrict__ t_eval,
                      float* __restrict__ out,
                      int B, int n_steps)
{
  extern __shared__ float lds[];            // [2][WG][WPAD] floats = 149504 B
  const int tid     = threadIdx.x;
  const int wgFirst = blockIdx.x * WG;
  const int b       = wgFirst + tid;

  // ---- Stage this workgroup's W_pos / W_neg slabs into LDS.
  // Coalesced gfx1250 async copy: consecutive lanes fetch consecutive dwords
  // straight into LDS (no VGPR round-trip), re-packed to a 292-dword padded
  // stride. NT hint: W is consumed exactly once from global.
  {
    int slabElems = B - wgFirst; if (slabElems > WG) slabElems = WG;
    const int totalDw = slabElems * NN;
    const uint32_t ldsBase = (uint32_t)(uintptr_t)(void*)lds;
    const uint32_t negOfs  = (uint32_t)(WG * WPAD) * 4u;
    uint32_t gOff = (uint32_t)(wgFirst * NN + tid) * 4u;
    int e = 0, r = tid;                    // d = e*NN + r, incremental carry
    for (int d = tid; d < totalDw; d += WG) {
      const uint32_t dstP = ldsBase + (uint32_t)(e * WPAD + r) * 4u;
      asm volatile("global_load_async_to_lds_b32 %0, %1, %2 th:TH_LOAD_NT"
                   :: "v"(dstP), "v"(gOff), "s"(W_pos) : "memory");
      asm volatile("global_load_async_to_lds_b32 %0, %1, %2 th:TH_LOAD_NT"
                   :: "v"(dstP + negOfs), "v"(gOff), "s"(W_neg) : "memory");
      gOff += (uint32_t)WG * 4u;
      r += WG;
      if (r >= NN) { r -= NN; ++e; }
    }
    asm volatile("s_wait_asynccnt 0x0" ::: "memory");
  }
  __syncthreads();

  if (b >= B) return;

  // ---- Per-element constants in registers
  float s[NDIM], Pv[NDIM], reluP[9], reluNP[9], fz[4];
#pragma unroll
  for (int i = 0; i < NDIM; ++i) {
    s[i]  = state0[(size_t)b * NDIM + i];
    Pv[i] = pert  [(size_t)b * NDIM + i];
  }
#pragma unroll
  for (int i = 0; i < 9; ++i) {
    reluP [i] = fmaxf( Pv[i], 0.0f);
    reluNP[i] = fmaxf(-Pv[i], 0.0f);
  }
#pragma unroll
  for (int k = 0; k < 4; ++k) fz[k] = feas[(size_t)b * 4 + k];

  const float dt = t_eval[1] - t_eval[0];

  // t = 0 row is state0 verbatim
#pragma unroll
  for (int i = 0; i < NDIM; ++i) out[(size_t)b * NDIM + i] = s[i];

  const float* __restrict__ Wp = &lds[tid * WPAD];
  const float* __restrict__ Wn = &lds[WG * WPAD + tid * WPAD];

  const size_t rowStride = (size_t)B * NDIM;
  float* __restrict__ outPtr = out + rowStride + (size_t)b * NDIM;

#pragma unroll 1
  for (int t = 0; t < n_steps; ++t) {
    float exc[NDIM], inh[NDIM];
#pragma unroll
    for (int i = 0; i < NDIM; ++i) { exc[i] = 0.0f; inh[i] = 0.0f; }

    // Two 17x17 matvecs, flat-indexed in 16B LDS chunks; (i,j) fold at compile time.
#pragma unroll
    for (int c = 0; c < NN / 4; ++c) {          // 72 chunks of 4 dwords
      const float4 wp4 = *(const float4*)(Wp + 4 * c);
      const float4 wn4 = *(const float4*)(Wn + 4 * c);
      const float wp[4] = {wp4.x, wp4.y, wp4.z, wp4.w};
      const float wn[4] = {wn4.x, wn4.y, wn4.z, wn4.w};
#pragma unroll
      for (int u = 0; u < 4; ++u) {
        const int idx = 4 * c + u;
        const int i = idx / NDIM;
        const int j = idx - i * NDIM;
        exc[i] = fmaf(wp[u], s[j], exc[i]);
        inh[i] = fmaf(wn[u], s[j], inh[i]);
      }
    }
    exc[16] = fmaf(Wp[NN - 1], s[16], exc[16]);  // idx 288 -> (16,16)
    inh[16] = fmaf(Wn[NN - 1], s[16], inh[16]);

    // Gates on value units
    float ge[4], gi[4];
#pragma unroll
    for (int k = 0; k < 4; ++k) {
      const float ve = s[13 + k] + Pv[13 + k];
      ge[k] = sigmoid_f( 1.5f  * ve);
      gi[k] = sigmoid_f(-0.75f * ve);
    }

    float E[NDIM], I[NDIM];
#pragma unroll
    for (int i = 0; i < NDIM; ++i) {
      const bool act = (i >= 9 && i < 13);
      const float gE = act ? ge[i - 9] : 1.0f;
      const float gI = act ? gi[i - 9] : 1.0f;
      E[i] = fmaxf(gE * exc[i], 0.0f);
      I[i] = fmaxf(gI * inh[i], 0.0f);
    }
#pragma unroll
    for (int i = 0; i < 9; ++i) { E[i] += reluP[i]; I[i] += reluNP[i]; }

    // Lateral inhibition among activity units (uses pre-update state)
    const float asum = s[9] + s[10] + s[11] + s[12];
#pragma unroll
    for (int k = 0; k < 4; ++k) {
      const float os  = asum - s[9 + k];
      const float lat = 3.0f * os * fast_rcp(0.3f + os);
      E[9 + k] *= fz[k];
      I[9 + k] += lat;
    }

    // Shunting update + clip + store trajectory row
#pragma unroll
    for (int i = 0; i < NDIM; ++i) {
      const float pd   = (i >= 13) ? Pv[i] : 0.0f;
      const float dsdt = (-0.15f * s[i] + (1.0f - s[i]) * E[i]
                          - (0.1f + s[i]) * I[i] + pd) * (1.0f / 0.8f);
      float v = fmaf(dt, dsdt, s[i]);
      const float lo = (i >= 13) ? -1.0f : 0.0f;
      v = fminf(fmaxf(v, lo), 1.0f);
      s[i] = v;
      outPtr[i] = v;
    }
    outPtr += rowStride;
  }
}

extern "C" void kernel_launch(void* const* d_in, const int* in_sizes, int n_in,
                              void* d_out, int out_size, void* d_ws, size_t ws_size,
                              hipStream_t stream) {
  const float* state0 = (const float*)d_in[0];
  const float* W_pos  = (const float*)d_in[1];
  const float* W_neg  = (const float*)d_in[2];
  const float* feas   = (const float*)d_in[3];
  const float* pert   = (const float*)d_in[4];
  const float* t_eval = (const float*)d_in[5];
  float* out = (float*)d_out;

  const int B       = in_sizes[0] / NDIM;   // 32768
  const int n_steps = in_sizes[5] - 1;      // 127
  const int grid    = (B + WG - 1) / WG;
  const size_t shmem = (size_t)2 * WG * WPAD * sizeof(float);

  grossberg_scan_kernel<<<grid, WG, shmem, stream>>>(
      state0, W_pos, W_neg, feas, pert, t_eval, out, B, n_steps);
}